// FlaxMambaMixer_16423954940326
// MI455X (gfx1250) — compile-verified
//
#include <hip/hip_runtime.h>
#include <hip/hip_bf16.h>

// Problem constants (from reference)
constexpr int Lc = 2048;   // sequence length
constexpr int Hc = 2048;   // hidden
constexpr int Dc = 4096;   // intermediate
constexpr int Nc = 16;     // state size
constexpr int Rc = 128;    // dt rank
constexpr int Kc = 4;      // conv kernel

typedef __attribute__((ext_vector_type(2))) float v2f;
typedef __attribute__((ext_vector_type(8))) float v8f;

// Async global->LDS copy of 16 bytes per lane (CDNA5 GLOBAL_LOAD_ASYNC_TO_LDS_B128,
// tracked by ASYNCcnt). LDS address = low 32 bits of the flat shared-memory
// address (hardware truncates the LDS aperture address to the LDS offset).
__device__ __forceinline__ void async_copy16(const float* g, float* l) {
  const unsigned int loff = (unsigned int)(size_t)l;
  asm volatile("global_load_async_to_lds_b128 %0, %1, off"
               :
               : "v"(loff), "v"(g)
               : "memory");
}

// ---------------------------------------------------------------------------
// Tiled f32 WMMA GEMM:  C[M,N] = A[M,K] @ B[K,N]   (A row stride = lda)
// Macro tile 128x128, BK=16, double-buffered LDS fed by async global->LDS
// copies (ASYNCcnt pipeline). 256 threads = 8 waves (2x4), each wave computes
// 64x32 = 4x2 subtiles of 16x16 via V_WMMA_F32_16X16X4_F32.
// EPI: 0 = plain store, 1 = softplus(acc + bias[col]) (dt projection)
// Requirements: M % 128 == 0, K % 16 == 0, N % 4 == 0 (N guarded vs 128).
// ---------------------------------------------------------------------------
template <int EPI>
__global__ __launch_bounds__(256) void gemm_wmma_f32(
    const float* __restrict__ A, const float* __restrict__ B,
    float* __restrict__ C, const float* __restrict__ bias,
    int M, int N, int K, int lda) {
  constexpr int BK = 16;
  constexpr int ALD = 20;  // A LDS row stride: 16B-aligned rows, conflict-free
  __shared__ float As[2][128 * ALD];   // [m][k] padded
  __shared__ float Bs[2][BK * 128];    // [k][n] natural layout

  const int tid = threadIdx.x;
  const int lane = tid & 31;
  const int wave = tid >> 5;
  const int waveM = wave >> 2;     // 0..1
  const int waveN = wave & 3;      // 0..3
  const int laneHalf = lane >> 4;  // selects K pair
  const int laneMod = lane & 15;   // m (A) or n (B) within subtile

  const int mBlock = blockIdx.y * 128;
  const int nBlock = blockIdx.x * 128;
  const int mWave = waveM * 64;
  const int nWave = waveN * 32;
  const bool nPartial = (nBlock + 128 > N);

  // Static per-thread copy coordinates: 2 A-chunks + 2 B-chunks of 16B each.
  const int aRow0 = tid >> 2;           // rows 0..63
  const int aKc = (tid & 3) << 2;       // 0,4,8,12
  const int aRow1 = aRow0 + 64;         // rows 64..127
  const int bKr0 = tid >> 5;            // k rows 0..7
  const int bNc = (tid & 31) << 2;      // n cols 0..124
  const int bKr1 = bKr0 + 8;            // k rows 8..15
  const int ngRaw = nBlock + bNc;
  const int ngClamped = (ngRaw + 3 < N) ? ngRaw : (N - 4);  // legal address;
                                                            // patched below

  auto issue = [&](int k0, int buf) {
    async_copy16(A + (size_t)(mBlock + aRow0) * lda + k0 + aKc,
                 &As[buf][aRow0 * ALD + aKc]);
    async_copy16(A + (size_t)(mBlock + aRow1) * lda + k0 + aKc,
                 &As[buf][aRow1 * ALD + aKc]);
    async_copy16(B + (size_t)(k0 + bKr0) * N + ngClamped,
                 &Bs[buf][bKr0 * 128 + bNc]);
    async_copy16(B + (size_t)(k0 + bKr1) * N + ngClamped,
                 &Bs[buf][bKr1 * 128 + bNc]);
  };

  v8f acc[4][2] = {};

  const int T = K / BK;
  issue(0, 0);
  for (int t = 0; t < T; ++t) {
    const int buf = t & 1;
    if (t + 1 < T) {
      issue((t + 1) * BK, buf ^ 1);
      // 4 copies of the NEXT tile outstanding; everything older has landed.
      asm volatile("s_wait_asynccnt 0x4" ::: "memory");
    } else {
      asm volatile("s_wait_asynccnt 0x0" ::: "memory");
    }
    __syncthreads();

    const float* Asc = As[buf];
    float* Bsc = Bs[buf];

    if (nPartial) {  // uniform branch per block: zero out-of-range B columns
      for (int c = tid; c < BK * 128; c += 256)
        if (nBlock + (c & 127) >= N) Bsc[c] = 0.f;
      __syncthreads();
    }

#pragma unroll
    for (int kk = 0; kk < BK; kk += 4) {
      const int ks = kk + laneHalf * 2;
      v2f afrag[4], bfrag[2];
#pragma unroll
      for (int i = 0; i < 4; ++i)
        afrag[i] = *(const v2f*)&Asc[(mWave + i * 16 + laneMod) * ALD + ks];
#pragma unroll
      for (int j = 0; j < 2; ++j) {
        const int n = nWave + j * 16 + laneMod;
        v2f b;
        b.x = Bsc[ks * 128 + n];
        b.y = Bsc[(ks + 1) * 128 + n];
        bfrag[j] = b;
      }
#pragma unroll
      for (int i = 0; i < 4; ++i)
#pragma unroll
        for (int j = 0; j < 2; ++j)
          acc[i][j] = __builtin_amdgcn_wmma_f32_16x16x4_f32(
              false, afrag[i], false, bfrag[j], (short)0, acc[i][j], false,
              false);
    }
    __syncthreads();
  }

  // ---- epilogue: C layout VGPR r -> row = base + (lane>=16 ? 8:0) + r ----
#pragma unroll
  for (int i = 0; i < 4; ++i) {
#pragma unroll
    for (int j = 0; j < 2; ++j) {
      const int col = nBlock + nWave + j * 16 + laneMod;
      if (col < N) {
        const int rowBase = mBlock + mWave + i * 16 + laneHalf * 8;
        const float bval = (EPI == 1) ? bias[col] : 0.f;
#pragma unroll
        for (int r = 0; r < 8; ++r) {
          float v = acc[i][j][r];
          if (EPI == 1) {
            v += bval;
            v = (v > 20.f) ? v : log1pf(__expf(v));  // softplus
          }
          C[(size_t)(rowBase + r) * N + col] = v;
        }
      }
    }
  }
}

// ---------------------------------------------------------------------------
// Depthwise causal conv1d (K=4) + bias + SiLU.
// hidden channel d lives at proj[l*2D + d]; output h[l*D + d].
// ---------------------------------------------------------------------------
__global__ __launch_bounds__(256) void conv_silu_kernel(
    const float* __restrict__ proj, const float* __restrict__ kern,
    const float* __restrict__ bias, float* __restrict__ h) {
  const int idx = blockIdx.x * 256 + threadIdx.x;  // over L*D
  const int d = idx & (Dc - 1);
  const int l = idx >> 12;  // /Dc
  float acc = bias[d];
#pragma unroll
  for (int k = 0; k < Kc; ++k) {
    const int ls = l - (Kc - 1) + k;
    const float x = (ls >= 0) ? proj[(size_t)ls * (2 * Dc) + d] : 0.f;
    acc = fmaf(x, kern[d * Kc + k], acc);
  }
  h[(size_t)l * Dc + d] = acc * (1.f / (1.f + __expf(-acc)));
}

// ---------------------------------------------------------------------------
// Selective scan. One lane per channel d; N=16 states in registers.
// B/C vectors (ssm cols 128..159) staged through LDS in chunks of 32 steps.
// Reads dt from dtp[l,d], overwrites same slot with y[l,d] (buffer reuse).
// ---------------------------------------------------------------------------
__global__ __launch_bounds__(256) void scan_kernel(
    const float* __restrict__ ssm,    // [L,160]
    float* __restrict__ dtp,          // [L,D] in: dt, out: y
    const float* __restrict__ h,      // [L,D]
    const float* __restrict__ proj,   // [L,2D], gate at col D+d
    const float* __restrict__ A_log,  // [D,16]
    const float* __restrict__ Dp)     // [D]
{
  __shared__ float bc[32 * 32];  // 32 timesteps x (16 B | 16 C)
  const int d = blockIdx.x * 256 + threadIdx.x;

  float Acoef[Nc];
#pragma unroll
  for (int n = 0; n < Nc; ++n) Acoef[n] = -__expf(A_log[d * Nc + n]);
  float s[Nc] = {};
  const float dpar = Dp[d];

  for (int l0 = 0; l0 < Lc; l0 += 32) {
    __syncthreads();
#pragma unroll
    for (int i = 0; i < 4; ++i) {
      const int idx = threadIdx.x + i * 256;
      const int r = idx >> 5, c = idx & 31;
      bc[idx] = ssm[(size_t)(l0 + r) * (Rc + 2 * Nc) + Rc + c];
    }
    __syncthreads();

    for (int li = 0; li < 32; ++li) {
      const size_t l = (size_t)(l0 + li);
      const float dt = dtp[l * Dc + d];
      const float hv = h[l * Dc + d];
      const float g = proj[l * (2 * Dc) + Dc + d];
      float y = 0.f;
#pragma unroll
      for (int n = 0; n < Nc; ++n) {
        const float dA = __expf(Acoef[n] * dt);
        s[n] = fmaf(dA, s[n], dt * bc[li * 32 + n] * hv);
        y = fmaf(s[n], bc[li * 32 + 16 + n], y);
      }
      y = fmaf(hv, dpar, y);
      const float sg = g * (1.f / (1.f + __expf(-g)));
      dtp[l * Dc + d] = y * sg;
    }
  }
}

// ---------------------------------------------------------------------------
extern "C" void kernel_launch(void* const* d_in, const int* in_sizes, int n_in,
                              void* d_out, int out_size, void* d_ws,
                              size_t ws_size, hipStream_t stream) {
  const float* input = (const float*)d_in[0];    // [L,H]
  const float* W_in = (const float*)d_in[1];     // [H,2D]
  const float* conv_k = (const float*)d_in[2];   // [D,1,K]
  const float* conv_b = (const float*)d_in[3];   // [D]
  const float* W_x = (const float*)d_in[4];      // [D,R+2N]
  const float* W_dt = (const float*)d_in[5];     // [R,D]
  const float* b_dt = (const float*)d_in[6];     // [D]
  const float* W_out = (const float*)d_in[7];    // [D,H]
  const float* A_log = (const float*)d_in[8];    // [D,N]
  const float* D_param = (const float*)d_in[9];  // [D]
  float* out = (float*)d_out;                    // [L,H]

  float* ws = (float*)d_ws;
  float* proj = ws;                               // [L, 2D]
  float* h = proj + (size_t)Lc * 2 * Dc;          // [L, D]
  float* ssm = h + (size_t)Lc * Dc;               // [L, 160]
  float* dtp = ssm + (size_t)Lc * (Rc + 2 * Nc);  // [L, D] (dt, then y)

  const dim3 blk(256);
  const int NS = Rc + 2 * Nc;  // 160

  // 1) in_proj: proj[L,2D] = input[L,H] @ W_in[H,2D]
  gemm_wmma_f32<0><<<dim3((2 * Dc) / 128, Lc / 128), blk, 0, stream>>>(
      input, W_in, proj, nullptr, Lc, 2 * Dc, Hc, Hc);

  // 2) depthwise conv + bias + SiLU -> h[L,D]
  conv_silu_kernel<<<(Lc * Dc) / 256, blk, 0, stream>>>(proj, conv_k, conv_b, h);

  // 3) x_proj: ssm[L,160] = h[L,D] @ W_x[D,160]
  gemm_wmma_f32<0><<<dim3((NS + 127) / 128, Lc / 128), blk, 0, stream>>>(
      h, W_x, ssm, nullptr, Lc, NS, Dc, Dc);

  // 4) dt: dtp[L,D] = softplus(ssm[:, :R] @ W_dt[R,D] + b_dt)
  gemm_wmma_f32<1><<<dim3(Dc / 128, Lc / 128), blk, 0, stream>>>(
      ssm, W_dt, dtp, b_dt, Lc, Dc, Rc, NS);

  // 5) selective scan (+ D skip + gate) -> y in dtp
  scan_kernel<<<Dc / 256, blk, 0, stream>>>(ssm, dtp, h, proj, A_log, D_param);

  // 6) out_proj: out[L,H] = y[L,D] @ W_out[D,H]
  gemm_wmma_f32<0><<<dim3(Hc / 128, Lc / 128), blk, 0, stream>>>(
      dtp, W_out, out, nullptr, Lc, Hc, Dc, Dc);
}